// SoftDTW_76553497084652
// MI455X (gfx1250) — compile-verified
//
#include <hip/hip_runtime.h>

// ---------------------------------------------------------------------------
// Soft-DTW on MI455X (gfx1250, wave32)
//   Phase 1: batched cost matrix D = x2 + y2 - 2*X*Y^T via V_WMMA_F32_16X16X4_F32
//   Phase 2: anti-diagonal wavefront DP, 2x2-cell tiles per thread (halves the
//            serial barrier count: 511 super-steps instead of 1023)
// Shapes fixed by the reference: B=64, N=M=512, K(feature)=256, gamma=1.
// ---------------------------------------------------------------------------

typedef float v2f __attribute__((ext_vector_type(2)));
typedef float v8f __attribute__((ext_vector_type(8)));

#define B_   64
#define N_   512
#define M_   512
#define K_   256
#define LARGE_ 1e10f

// ---------------------------------------------------------------------------
// Kernel 1: squared row norms for X and Y.
// ---------------------------------------------------------------------------
__global__ __launch_bounds__(256) void sdtw_norms(const float* __restrict__ X,
                                                  const float* __restrict__ Y,
                                                  float* __restrict__ x2,
                                                  float* __restrict__ y2) {
    int t = blockIdx.x * blockDim.x + threadIdx.x;
    const int half = B_ * N_;
    const float* src;
    float* dst;
    int row;
    if (t < half) { src = X; dst = x2; row = t; }
    else          { src = Y; dst = y2; row = t - half; }
    const float4* p = (const float4*)(src + (size_t)row * K_);
    float s = 0.0f;
#pragma unroll 8
    for (int c = 0; c < K_ / 4; ++c) {
        float4 v = p[c];
        s += v.x * v.x + v.y * v.y + v.z * v.z + v.w * v.w;
    }
    dst[row] = s;
}

// ---------------------------------------------------------------------------
// Kernel 2: cost-matrix GEMM with WMMA f32 16x16x4 (unchanged; emits 64 wmma
// per wave with single-buffered global->LDS->wmma flow).
// ---------------------------------------------------------------------------
#define BM 64
#define BN 32
#define BK 32
#define LDS_STRIDE 36

__global__ __launch_bounds__(256) void sdtw_cost_gemm(const float* __restrict__ X,
                                                      const float* __restrict__ Y,
                                                      const float* __restrict__ x2,
                                                      const float* __restrict__ y2,
                                                      float* __restrict__ D) {
    __shared__ float Xs[BM][LDS_STRIDE];
    __shared__ float Ys[BN][LDS_STRIDE];

    const int b  = blockIdx.z;
    const int m0 = blockIdx.y * BM;
    const int n0 = blockIdx.x * BN;

    const int tid  = threadIdx.x;
    const int wave = tid >> 5;
    const int lane = tid & 31;
    const int wm   = wave & 3;
    const int wn   = wave >> 2;
    const int lr   = lane & 15;
    const int hk   = lane >> 4;

    const float* Xb = X + ((size_t)b * N_) * K_;
    const float* Yb = Y + ((size_t)b * M_) * K_;

    v8f acc = {};

    for (int kk = 0; kk < K_; kk += BK) {
        __syncthreads();
#pragma unroll
        for (int it = 0; it < 2; ++it) {
            int idx = tid + it * 256;
            int r  = idx >> 3;
            int c4 = (idx & 7) << 2;
            float4 v = *(const float4*)(Xb + (size_t)(m0 + r) * K_ + kk + c4);
            *(float4*)&Xs[r][c4] = v;
        }
        {
            int r  = tid >> 3;
            int c4 = (tid & 7) << 2;
            float4 v = *(const float4*)(Yb + (size_t)(n0 + r) * K_ + kk + c4);
            *(float4*)&Ys[r][c4] = v;
        }
        __syncthreads();

#pragma unroll
        for (int k4 = 0; k4 < BK; k4 += 4) {
            v2f a  = *(const v2f*)&Xs[wm * 16 + lr][k4 + hk * 2];
            v2f bb = *(const v2f*)&Ys[wn * 16 + lr][k4 + hk * 2];
            acc = __builtin_amdgcn_wmma_f32_16x16x4_f32(
                false, a, false, bb, (short)0, acc, false, false);
        }
    }

    const int col = n0 + wn * 16 + lr;
    const float y2v = y2[(size_t)b * M_ + col];
    float* Dtile = D + (size_t)b * N_ * M_;
#pragma unroll
    for (int v = 0; v < 8; ++v) {
        int row = m0 + wm * 16 + hk * 8 + v;
        float val = x2[(size_t)b * N_ + row] + y2v - 2.0f * acc[v];
        Dtile[(size_t)row * M_ + col] = val;
    }
}

// ---------------------------------------------------------------------------
// Kernel 3: soft-DTW wavefront DP with 2x2 tiles.
// Thread i (0..255) owns rows r0=2i, r1=2i+1. Super-step s computes tile
// q = s - i (cols c0=2q, c1=2q+1) when 0 <= q < 256.  511 super-steps total.
// Boundary exchange per tile via LDS ring (3 phases):
//   B [ph][i] = {R[r1][c0], R[r1][c1]}   (bottom row -> thread i+1)
//   Rt[ph][i] = {R[r0][c1], R[r1][c1]}   (right col  -> same thread, next q)
// gamma = 1: softmin(a,b,c) = m - log(exp(m-a)+exp(m-b)+exp(m-c)).
// ---------------------------------------------------------------------------
__device__ __forceinline__ float softmin3(float a, float b, float c) {
    float m = fminf(a, fminf(b, c));
    float e = __expf(m - a) + __expf(m - b) + __expf(m - c);
    return m - __logf(e);
}

__global__ __launch_bounds__(256) void sdtw_dp(const float* __restrict__ D,
                                               float* __restrict__ out) {
    __shared__ float Bb[3][N_ / 2][2];   // bottom-row outputs
    __shared__ float Rr[3][N_ / 2][2];   // right-col outputs

    const int b = blockIdx.x;
    const int i = threadIdx.x;           // tile-row (rows 2i, 2i+1)
    const float* Db = D + (size_t)b * N_ * M_;

    // Phases for s-1 and s-2 start as virtual boundary (LARGE).
    Bb[1][i][0] = LARGE_; Bb[1][i][1] = LARGE_;
    Bb[2][i][0] = LARGE_; Bb[2][i][1] = LARGE_;
    Rr[1][i][0] = LARGE_; Rr[1][i][1] = LARGE_;
    Rr[2][i][0] = LARGE_; Rr[2][i][1] = LARGE_;
    __syncthreads();

    float last = LARGE_;
    const int QT = M_ / 2;               // 256 tile-cols
    for (int s = 0; s < N_ / 2 + QT - 1; ++s) {   // 511 super-steps
        const int cur = s % 3;
        const int prv = (s + 2) % 3;     // s-1
        const int pp  = (s + 1) % 3;     // s-2

        const int q = s - i;
        const bool valid = (q >= 0) && (q < QT);

        float c01 = LARGE_, c10 = LARGE_, c11 = LARGE_;
        if (valid) {
            // Gather boundary values.
            float up0, up1, lf0, lf1, dg;
            if (i > 0) { up0 = Bb[prv][i - 1][0]; up1 = Bb[prv][i - 1][1]; }
            else       { up0 = LARGE_;            up1 = LARGE_; }
            if (q > 0) { lf0 = Rr[prv][i][0];     lf1 = Rr[prv][i][1]; }
            else       { lf0 = LARGE_;            lf1 = LARGE_; }
            if (i > 0 && q > 0)       dg = Bb[pp][i - 1][1];
            else if (i == 0 && q == 0) dg = 0.0f;       // R[-1][-1]
            else                       dg = LARGE_;

            // Cost values: rows 2i,2i+1 x cols 2q,2q+1 as two b64 loads.
            float2 d0 = *(const float2*)(Db + (size_t)(2 * i)     * M_ + 2 * q);
            float2 d1 = *(const float2*)(Db + (size_t)(2 * i + 1) * M_ + 2 * q);

            // 2x2 cells in dependency order.
            float c00 = d0.x + softmin3(dg,  up0, lf0);
            c01       = d0.y + softmin3(up0, up1, c00);
            c10       = d1.x + softmin3(lf0, c00, lf1);
            c11       = d1.y + softmin3(c00, c01, c10);
            last = c11;
        }

        // Publish boundaries (LARGE when invalid keeps ghosts consistent).
        Bb[cur][i][0] = c10; Bb[cur][i][1] = c11;
        Rr[cur][i][0] = c01; Rr[cur][i][1] = c11;
        __syncthreads();
    }

    // R[N-1][M-1] = c11 of tile (255,255), produced by thread 255 at s=510.
    if (i == N_ / 2 - 1) out[b] = last;
}

// ---------------------------------------------------------------------------
// Launch wrapper.
//   d_in[0] = X (B*N*K f32), d_in[1] = Y (B*M*K f32); d_out = (B,) f32
//   d_ws: D (B*N*M f32) | x2 (B*N f32) | y2 (B*M f32)  ~= 67.4 MB
// ---------------------------------------------------------------------------
extern "C" void kernel_launch(void* const* d_in, const int* in_sizes, int n_in,
                              void* d_out, int out_size, void* d_ws, size_t ws_size,
                              hipStream_t stream) {
    const float* X = (const float*)d_in[0];
    const float* Y = (const float*)d_in[1];
    float* out = (float*)d_out;

    float* D  = (float*)d_ws;
    float* x2 = D + (size_t)B_ * N_ * M_;
    float* y2 = x2 + (size_t)B_ * N_;

    {
        int total = 2 * B_ * N_;
        sdtw_norms<<<(total + 255) / 256, 256, 0, stream>>>(X, Y, x2, y2);
    }
    {
        dim3 grid(M_ / BN, N_ / BM, B_);
        sdtw_cost_gemm<<<grid, 256, 0, stream>>>(X, Y, x2, y2, D);
    }
    {
        sdtw_dp<<<B_, N_ / 2, 0, stream>>>(D, out);
    }
}